// FClip_3289944948920
// MI455X (gfx1250) — compile-verified
//
#include <hip/hip_runtime.h>
#include <hip/hip_bf16.h>
#include <math.h>

// ---------------------------------------------------------------------------
// FClip post-processing for MI455X (gfx1250, wave32).
// Pairwise structural-NMS distance matrix is computed with
// V_WMMA_F32_16X16X4_F32 (K=4 matches the 4-coordinate line descriptor).
// ---------------------------------------------------------------------------

typedef __attribute__((ext_vector_type(2))) float v2f;
typedef __attribute__((ext_vector_type(8))) float v8f;

#define HDIM 1024
#define WDIM 1024
#define HWSZ (HDIM * WDIM)
#define KMAX 1024
#define CANDMAX 2048

// workspace byte offsets
#define WS_SMAPU  0u                      // HWSZ * 4        = 4,194,304
#define WS_HIST   4194304u                // 256 * 4
#define WS_STATE  4195328u                // 16 * 4
#define WS_CAND   4195392u                // 2048 * 8
#define WS_TOPIDX 4211776u                // 1024 * 4
#define WS_P      4215872u                // 1024 * 16
#define WS_NORM   4232256u                // 1024 * 4
#define WS_IND    4236352u                // 1024 * 32 * 4   = 131,072

__device__ __forceinline__ float sigm(float t) { return 1.0f / (1.0f + expf(-t)); }

// monotone float<->uint mapping (descending float == descending uint)
__device__ __forceinline__ unsigned f2ord(float f) {
    unsigned u = __float_as_uint(f);
    return (u & 0x80000000u) ? ~u : (u | 0x80000000u);
}
__device__ __forceinline__ float ord2f(unsigned u) {
    return (u & 0x80000000u) ? __uint_as_float(u & 0x7FFFFFFFu) : __uint_as_float(~u);
}

// --- 1. fused softmax-prob + 3x3 soft-NMS + ordinal key (one read of hm[0:2])
__global__ void fclip_smap(const float* __restrict__ hm, unsigned* __restrict__ smapU) {
    int gid = blockIdx.x * blockDim.x + threadIdx.x;
    if (gid >= HWSZ) return;
    int y = gid >> 10, x = gid & 1023;
    const float* h0 = hm;
    const float* h1 = hm + HWSZ;
    float c = sigm(h1[gid] - h0[gid]);  // softmax([h0,h1])[1]
    float mx = c;
#pragma unroll
    for (int dy = -1; dy <= 1; ++dy)
#pragma unroll
        for (int dx = -1; dx <= 1; ++dx) {
            int yy = y + dy, xx = x + dx;
            if (yy >= 0 && yy < HDIM && xx >= 0 && xx < WDIM) {
                int q = yy * WDIM + xx;
                mx = fmaxf(mx, sigm(h1[q] - h0[q]));
            }
        }
    float s = (c == mx) ? c : c * 0.8f;  // SOFT = 0.8
    smapU[gid] = f2ord(s);
}

// --- 2. radix-select state init
__global__ void fclip_init(unsigned* state, unsigned* hist, const int* kptr) {
    state[0] = 0u;                   // prefix
    state[1] = (unsigned)(*kptr);    // remaining rank
    state[2] = 0u;                   // kth ordinal value
    state[3] = 0u;                   // candidate count
    for (int i = 0; i < 256; ++i) hist[i] = 0u;
}

// --- 3a. per-pass byte histogram of prefix-matching keys
__global__ void fclip_count(const unsigned* __restrict__ smapU,
                            const unsigned* __restrict__ state,
                            unsigned* __restrict__ hist, int s) {
    __shared__ unsigned lh[256];
    int t = threadIdx.x;
    lh[t] = 0u;
    __syncthreads();
    unsigned prefix = state[0];
    unsigned pmask = (s == 24) ? 0u : (0xFFFFFFFFu << (s + 8));
    int stride = gridDim.x * blockDim.x;
    for (int i = blockIdx.x * blockDim.x + t; i < HWSZ; i += stride) {
        unsigned u = smapU[i];
        if (((u ^ prefix) & pmask) == 0u) atomicAdd(&lh[(u >> s) & 255u], 1u);
    }
    __syncthreads();
    if (lh[t]) atomicAdd(&hist[t], lh[t]);
}

// --- 3b. single-thread scan: pick byte bin containing the k-th largest
__global__ void fclip_scan(unsigned* hist, unsigned* state, int s) {
    unsigned r = state[1];
    unsigned prefix = state[0];
    unsigned cum = 0u;
    for (int b = 255; b >= 0; --b) {
        unsigned c = hist[b];
        if (cum + c >= r) { prefix |= ((unsigned)b) << s; r -= cum; break; }
        cum += c;
    }
    state[0] = prefix;
    state[1] = r;
    if (s == 0) state[2] = prefix;   // exact ordinal of k-th largest
    for (int i = 0; i < 256; ++i) hist[i] = 0u;  // clear for next pass
}

// --- 4. compact all keys >= kth ordinal
__global__ void fclip_compact(const unsigned* __restrict__ smapU,
                              unsigned* __restrict__ state,
                              uint2* __restrict__ cand) {
    unsigned kth = state[2];
    int stride = gridDim.x * blockDim.x;
    for (int i = blockIdx.x * blockDim.x + threadIdx.x; i < HWSZ; i += stride) {
        unsigned u = smapU[i];
        if (u >= kth) {
            unsigned pos = atomicAdd(&state[3], 1u);
            if (pos < CANDMAX) cand[pos] = make_uint2(u, (unsigned)i);
        }
    }
}

// --- 5. single-block bitonic sort, key = (ordinal << 32) | ~idx (ties -> smaller idx)
__global__ void __launch_bounds__(1024) fclip_sort(const uint2* __restrict__ cand,
                                                   const unsigned* __restrict__ state,
                                                   unsigned* __restrict__ topIdx,
                                                   float* __restrict__ out,
                                                   const int* kptr) {
    __shared__ unsigned long long keys[CANDMAX];
    int t = threadIdx.x;
    unsigned cnt = state[3];
    if (cnt > CANDMAX) cnt = CANDMAX;
    for (int j = t; j < CANDMAX; j += 1024) {
        unsigned long long key = 0ull;
        if ((unsigned)j < cnt) {
            uint2 e = cand[j];
            key = ((unsigned long long)e.x << 32) | (unsigned long long)(~e.y);
        }
        keys[j] = key;
    }
    __syncthreads();
    for (unsigned ks = 2; ks <= CANDMAX; ks <<= 1) {
        for (unsigned js = ks >> 1; js > 0; js >>= 1) {
            for (unsigned i = t; i < CANDMAX; i += 1024) {
                unsigned ixj = i ^ js;
                if (ixj > i) {
                    bool desc = ((i & ks) == 0);
                    unsigned long long A = keys[i], B = keys[ixj];
                    if (desc ? (A < B) : (A > B)) { keys[i] = B; keys[ixj] = A; }
                }
            }
            __syncthreads();
        }
    }
    int n = *kptr;
    if (t < n && t < KMAX) {
        unsigned long long key = keys[t];
        unsigned idx = ~((unsigned)(key & 0xFFFFFFFFull));
        unsigned u = (unsigned)(key >> 32);
        topIdx[t] = idx;
        out[4 * n + t] = ord2f(u);   // scores section
    }
}

// --- 6. gather + trig -> lines [k,4]; also padded P and row squared-norms
__global__ void fclip_build(const float* __restrict__ hm,
                            const unsigned* __restrict__ topIdx,
                            float* __restrict__ out, float4* __restrict__ P,
                            float* __restrict__ norms, const int* kptr) {
    int j = blockIdx.x * blockDim.x + threadIdx.x;
    if (j >= KMAX) return;
    int n = *kptr;
    if (j < n) {
        unsigned idx = topIdx[j];
        float x = (float)(idx & 1023u);
        float y = (float)(idx >> 10);
        float off0 = sigm(hm[3 * HWSZ + idx]);  // coff channel swap [1,0]
        float off1 = sigm(hm[2 * HWSZ + idx]);
        float cx = x + off0, cy = y + off1;
        float r = sigm(hm[4 * HWSZ + idx]) * 64.0f;          // RADIUS_SCALE
        float ang = sigm(hm[5 * HWSZ + idx]) * 3.14159265358979323846f;
        float dx = cosf(ang) * r;
        float dy = -fabsf(sinf(ang)) * r;
        float4 ln = make_float4(cx + dx, cy + dy, cx - dx, cy - dy);
        out[j * 4 + 0] = ln.x; out[j * 4 + 1] = ln.y;
        out[j * 4 + 2] = ln.z; out[j * 4 + 3] = ln.w;
        P[j] = ln;
        norms[j] = ln.x * ln.x + ln.y * ln.y + ln.z * ln.z + ln.w * ln.w;
    } else {
        P[j] = make_float4(0.f, 0.f, 0.f, 0.f);
        norms[j] = 0.f;
    }
}

__global__ void fclip_zero(unsigned* ind) {
    int i = blockIdx.x * blockDim.x + threadIdx.x;
    if (i < KMAX * 32) ind[i] = 0u;
}

// --- 7. pairwise distance via V_WMMA_F32_16X16X4_F32, one wave per 16x16 tile.
//  d(i,j) = min(|Pi|^2+|Qj|^2-2*Pi.Qj , ... with endpoint-swapped Qj) <= 2.0
__global__ void fclip_pair(const float4* __restrict__ P,
                           const float* __restrict__ norms,
                           unsigned* __restrict__ ind, const int* kptr) {
    int n = *kptr;
    int tJ = blockIdx.x, tI = blockIdx.y;
    int I0 = tI * 16, J0 = tJ * 16;
    int L = threadIdx.x;            // 0..31, full wave, EXEC all ones
    int m = L & 15, kh = L >> 4;
    float4 p = P[I0 + m];
    float4 q = P[J0 + m];
    // A 16x4 f32 layout: VGPR0 = K0 (lanes 0-15) / K2 (16-31); VGPR1 = K1/K3
    v2f a, b, bs;
    a.x = kh ? p.z : p.x;  a.y = kh ? p.w : p.y;
    b.x = kh ? q.z : q.x;  b.y = kh ? q.w : q.y;
    bs.x = kh ? q.x : q.z; bs.y = kh ? q.y : q.w;   // endpoint-swapped B
    v8f cz = {};
    v8f s1 = __builtin_amdgcn_wmma_f32_16x16x4_f32(false, a, false, b,  (short)0, cz, false, false);
    v8f s2 = __builtin_amdgcn_wmma_f32_16x16x4_f32(false, a, false, bs, (short)0, cz, false, false);
    int gj = J0 + m;                 // D column for this lane
    float nq = norms[gj];
    unsigned word = (unsigned)(J0 >> 5);
    unsigned shift = (tJ & 1) ? 16u : 0u;
#pragma unroll
    for (int v = 0; v < 8; ++v) {
        int gi = I0 + v + 8 * kh;    // D row: VGPR v, lane half kh
        float np = norms[gi];
        float d1 = np + nq - 2.0f * s1[v];
        float d2 = np + nq - 2.0f * s2[v];
        float d = fminf(d1, d2);
        bool bit = (d <= 2.0f) && (gi != gj) && (gi < n) && (gj < n);
        unsigned bal = __builtin_amdgcn_ballot_w32(bit);
        // bits[15:0] -> row I0+v, bits[31:16] -> row I0+v+8 (cols J0..J0+15)
        if (L == 0 && (bal & 0xFFFFu))
            atomicOr(&ind[(unsigned)(I0 + v) * 32u + word], (bal & 0xFFFFu) << shift);
        if (L == 16 && (bal >> 16))
            atomicOr(&ind[(unsigned)(I0 + v + 8) * 32u + word], (bal >> 16) << shift);
    }
}

// --- 8. sequential greedy suppression: one wave32, lane w owns drop word w
__global__ void fclip_greedy(const unsigned* __restrict__ ind, const int* kptr,
                             float* __restrict__ out) {
    int n = *kptr;
    int w = threadIdx.x;             // 0..31
    unsigned drop = ind[w];          // drop = ind[0]
    for (int i = 1; i < n - 2; ++i) {  // fori_loop(1, n-2)
        unsigned dw = __shfl(drop, i >> 5, 32);
        if (!((dw >> (i & 31)) & 1u)) {          // uniform branch
            unsigned row = ind[i * 32 + w];
            int iw = i >> 5;
            unsigned fmask;
            if (w < iw)      fmask = 0u;
            else if (w > iw) fmask = 0xFFFFFFFFu;
            else fmask = ((i & 31) == 31) ? 0u : (0xFFFFFFFFu << ((i & 31) + 1));
            drop |= row & fmask;     // j > i
        }
    }
    float* keep = out + 5 * n;
#pragma unroll
    for (int b = 0; b < 32; ++b) {
        int j = w * 32 + b;
        if (j < n) keep[j] = ((drop >> b) & 1u) ? 0.0f : 1.0f;
    }
}

extern "C" void kernel_launch(void* const* d_in, const int* in_sizes, int n_in,
                              void* d_out, int out_size, void* d_ws, size_t ws_size,
                              hipStream_t stream) {
    const float* hm = (const float*)d_in[0];   // [6,1024,1024] f32
    const int* kptr = (const int*)d_in[1];     // k == 1000
    float* out = (float*)d_out;                // lines[4k] | scores[k] | keep[k]
    char* ws = (char*)d_ws;

    unsigned* smapU  = (unsigned*)(ws + WS_SMAPU);
    unsigned* hist   = (unsigned*)(ws + WS_HIST);
    unsigned* state  = (unsigned*)(ws + WS_STATE);
    uint2*    cand   = (uint2*)(ws + WS_CAND);
    unsigned* topIdx = (unsigned*)(ws + WS_TOPIDX);
    float4*   P      = (float4*)(ws + WS_P);
    float*    norms  = (float*)(ws + WS_NORM);
    unsigned* ind    = (unsigned*)(ws + WS_IND);

    fclip_smap<<<HWSZ / 256, 256, 0, stream>>>(hm, smapU);
    fclip_init<<<1, 1, 0, stream>>>(state, hist, kptr);
    for (int s = 24; s >= 0; s -= 8) {
        fclip_count<<<1024, 256, 0, stream>>>(smapU, state, hist, s);
        fclip_scan<<<1, 1, 0, stream>>>(hist, state, s);
    }
    fclip_compact<<<1024, 256, 0, stream>>>(smapU, state, cand);
    fclip_sort<<<1, 1024, 0, stream>>>(cand, state, topIdx, out, kptr);
    fclip_build<<<KMAX / 256, 256, 0, stream>>>(hm, topIdx, out, P, norms, kptr);
    fclip_zero<<<(KMAX * 32) / 256, 256, 0, stream>>>(ind);
    fclip_pair<<<dim3(64, 64), 32, 0, stream>>>(P, norms, ind, kptr);
    fclip_greedy<<<1, 32, 0, stream>>>(ind, kptr, out);
}